// BBBGraphConv_20598663152186
// MI455X (gfx1250) — compile-verified
//
#include <hip/hip_runtime.h>
#include <hip/hip_bf16.h>
#include <math.h>

typedef __attribute__((ext_vector_type(2))) float v2f;
typedef __attribute__((ext_vector_type(8))) float v8f;

#define N_NODES 100000
#define N_EDGES 1600000
#define CH 128
#define LDS_STRIDE 132  // 132 % 4 == 0 -> 16B-aligned rows; 132 % 64 = 4 -> conflict-free

// ---------------- workspace layout (in floats) ----------------
#define OFF_W     0                       // 128*128
#define OFF_BIAS  (OFF_W + CH*CH)         // 128
#define OFF_ODEG  (OFF_BIAS + CH)         // 100000
#define OFF_IDEG  (OFF_ODEG + N_NODES)    // 100000
#define OFF_H     (OFF_IDEG + N_NODES)    // 100000*128

// ---------------- zero a float buffer (float4 granularity) ----------------
__global__ void zero_f4(float* __restrict__ p, int n4) {
    int i = blockIdx.x * blockDim.x + threadIdx.x;
    if (i < n4) ((float4*)p)[i] = make_float4(0.f, 0.f, 0.f, 0.f);
}

// ---------------- Bayesian reparameterization: W = mu + eps*softplus(rho) ----------------
__global__ void prep_weights(const float* __restrict__ Wmu, const float* __restrict__ Wrho,
                             const float* __restrict__ Weps,
                             const float* __restrict__ bmu, const float* __restrict__ brho,
                             const float* __restrict__ beps,
                             float* __restrict__ W, float* __restrict__ bias) {
    int i = blockIdx.x * blockDim.x + threadIdx.x;
    if (i < CH * CH) {
        float sp = log1pf(expf(Wrho[i]));
        W[i] = Wmu[i] + Weps[i] * sp;
    } else if (i < CH * CH + CH) {
        int j = i - CH * CH;
        float sp = log1pf(expf(brho[j]));
        bias[j] = bmu[j] + beps[j] * sp;
    }
}

// ---------------- degree counts via f32 atomics ----------------
__global__ void degrees_kernel(const int* __restrict__ src, const int* __restrict__ dst,
                               float* __restrict__ odeg, float* __restrict__ ideg) {
    int e = blockIdx.x * blockDim.x + threadIdx.x;
    if (e < N_EDGES) {
        atomicAdd(&odeg[src[e]], 1.0f);
        atomicAdd(&ideg[dst[e]], 1.0f);
    }
}

// ---------------- h = (feat * odeg^-1/2) @ W via V_WMMA_F32_16X16X4_F32 ----------------
// 128 threads = 4 waves; block handles 16 rows x 128 cols; each wave does two 16x16 tiles.
__global__ __launch_bounds__(128) void gemm_wmma(const float* __restrict__ feat,
                                                 const float* __restrict__ odeg,
                                                 const float* __restrict__ W,
                                                 float* __restrict__ h) {
    __shared__ float lds[16 * LDS_STRIDE];
    const int t = threadIdx.x;
    const int row0 = blockIdx.x * 16;

    // Stage degree-scaled A slab (16 rows x 128 cols) into LDS.
    {
        const int r = t >> 3;            // 0..15
        const int c = (t & 7) * 16;      // 0,16,...,112
        const float s = rsqrtf(fmaxf(odeg[row0 + r], 1.0f));
        const float4* fp = (const float4*)&feat[(size_t)(row0 + r) * CH + c];
#pragma unroll
        for (int j = 0; j < 4; ++j) {
            float4 v = fp[j];
            v.x *= s; v.y *= s; v.z *= s; v.w *= s;
            *(float4*)&lds[r * LDS_STRIDE + c + j * 4] = v;
        }
    }
    __syncthreads();

    const int lane = t & 31;
    const int wave = t >> 5;
    const int half = lane >> 4;          // 0: K pair {k,k+1}; 1: K pair {k+2,k+3}
    const int l16  = lane & 15;
    const int n0 = wave * 16;            // tile 0 columns
    const int n1 = 64 + wave * 16;       // tile 1 columns

    v8f c0 = {};
    v8f c1 = {};
    const float* arow = &lds[l16 * LDS_STRIDE + half * 2];

#pragma unroll 4
    for (int k = 0; k < CH; k += 4) {
        v2f a;                            // A fragment: 16x4, rows=lanes%16, K split by half
        a.x = arow[k];
        a.y = arow[k + 1];

        const float* wk = &W[(size_t)(k + half * 2) * CH + l16];
        v2f b0, b1;                       // B fragment: 4x16, rows K / K+1 per VGPR
        b0.x = wk[n0];       b0.y = wk[CH + n0];
        b1.x = wk[n1];       b1.y = wk[CH + n1];

        c0 = __builtin_amdgcn_wmma_f32_16x16x4_f32(false, a, false, b0, (short)0, c0, false, false);
        c1 = __builtin_amdgcn_wmma_f32_16x16x4_f32(false, a, false, b1, (short)0, c1, false, false);
    }

    // C/D layout: VGPR r -> rows (r, r+8) across lane halves, N = lane%16
#pragma unroll
    for (int r = 0; r < 8; ++r) {
        const int row = row0 + r + half * 8;
        h[(size_t)row * CH + n0 + l16] = c0[r];
        h[(size_t)row * CH + n1 + l16] = c1[r];
    }
}

// ---------------- edge scatter: out[dst] += h[src], L2-resident atomics ----------------
// 32 threads per edge, each handles 4 channels (float4 gather + 4 f32 atomics).
__global__ void scatter_kernel(const int* __restrict__ src, const int* __restrict__ dst,
                               const float* __restrict__ h, float* __restrict__ out) {
    const int gid = blockIdx.x * blockDim.x + threadIdx.x;   // N_EDGES*32 = 51.2M < 2^31
    const int e = gid >> 5;
    if (e >= N_EDGES) return;
    const int q = (gid & 31) * 4;
    const int s = src[e];
    const int d = dst[e];
    const float4 v = *(const float4*)&h[(size_t)s * CH + q];
    float* o = &out[(size_t)d * CH + q];
    atomicAdd(o + 0, v.x);
    atomicAdd(o + 1, v.y);
    atomicAdd(o + 2, v.z);
    atomicAdd(o + 3, v.w);
}

// ---------------- finalize: out = out * ideg^-1/2 + bias ----------------
__global__ void finalize_kernel(const float* __restrict__ ideg, const float* __restrict__ bias,
                                float* __restrict__ out) {
    const int gid = blockIdx.x * blockDim.x + threadIdx.x;   // N_NODES*32
    if (gid >= N_NODES * 32) return;
    const int n = gid >> 5;
    const int q = (gid & 31) * 4;
    const float s = rsqrtf(fmaxf(ideg[n], 1.0f));
    float4 v = *(float4*)&out[(size_t)n * CH + q];
    const float4 b = *(const float4*)&bias[q];
    v.x = fmaf(v.x, s, b.x);
    v.y = fmaf(v.y, s, b.y);
    v.z = fmaf(v.z, s, b.z);
    v.w = fmaf(v.w, s, b.w);
    *(float4*)&out[(size_t)n * CH + q] = v;
}

extern "C" void kernel_launch(void* const* d_in, const int* in_sizes, int n_in,
                              void* d_out, int out_size, void* d_ws, size_t ws_size,
                              hipStream_t stream) {
    const float* feat  = (const float*)d_in[0];
    const int*   src   = (const int*)  d_in[1];
    const int*   dst   = (const int*)  d_in[2];
    const float* W_mu  = (const float*)d_in[3];
    const float* W_rho = (const float*)d_in[4];
    const float* b_mu  = (const float*)d_in[5];
    const float* b_rho = (const float*)d_in[6];
    const float* W_eps = (const float*)d_in[7];
    const float* b_eps = (const float*)d_in[8];

    float* ws   = (float*)d_ws;
    float* W    = ws + OFF_W;
    float* bias = ws + OFF_BIAS;
    float* odeg = ws + OFF_ODEG;
    float* ideg = ws + OFF_IDEG;
    float* h    = ws + OFF_H;
    float* out  = (float*)d_out;

    // 1) zero degree buffers (contiguous 200000 floats) and output (12.8M floats)
    {
        const int n4 = (2 * N_NODES) / 4;
        zero_f4<<<(n4 + 255) / 256, 256, 0, stream>>>(odeg, n4);
    }
    {
        const int n4 = (N_NODES * CH) / 4;
        zero_f4<<<(n4 + 255) / 256, 256, 0, stream>>>(out, n4);
    }

    // 2) reparameterized weights + bias
    prep_weights<<<(CH * CH + CH + 255) / 256, 256, 0, stream>>>(
        W_mu, W_rho, W_eps, b_mu, b_rho, b_eps, W, bias);

    // 3) degrees
    degrees_kernel<<<(N_EDGES + 255) / 256, 256, 0, stream>>>(src, dst, odeg, ideg);

    // 4) WMMA GEMM: h = (feat * odeg^-1/2) @ W   (100000 rows = 6250 x 16 exactly)
    gemm_wmma<<<N_NODES / 16, 128, 0, stream>>>(feat, odeg, W, h);

    // 5) edge scatter into out
    {
        const long long nthreads = (long long)N_EDGES * 32;
        scatter_kernel<<<(int)((nthreads + 255) / 256), 256, 0, stream>>>(src, dst, h, out);
    }

    // 6) finalize: dst-side norm + bias
    {
        const int nthreads = N_NODES * 32;
        finalize_kernel<<<(nthreads + 255) / 256, 256, 0, stream>>>(ideg, bias, out);
    }
}